// RCensNetConv_7911329759641
// MI455X (gfx1250) — compile-verified
//
#include <hip/hip_runtime.h>
#include <hip/hip_bf16.h>

typedef __attribute__((ext_vector_type(2))) float v2f;
typedef __attribute__((ext_vector_type(8))) float v8f;

#define NN   10000
#define EE   640000
#define RR   8
#define DD   128      // DIN == DOUT
#define EPSF 1e-8f

// ---------------------------------------------------------------------------
// deg[r][n] = EPS  (the +EPS from the reference is folded into the init)
// ---------------------------------------------------------------------------
__global__ void init_deg(float* __restrict__ deg) {
    int i = blockIdx.x * 256 + threadIdx.x;
    if (i < RR * NN) deg[i] = EPSF;
}

// ---------------------------------------------------------------------------
// Per edge: w_raw = mean(edge_features[e]); deg[r][dst] += |w_raw|;
// also copy edge_features into the 2nd output region (tuple return).
// ---------------------------------------------------------------------------
__global__ void edge_prep(const float* __restrict__ ef,
                          const int*   __restrict__ eidx,   // [2,E]: row0=src, row1=dst
                          const int*   __restrict__ etype,
                          float* __restrict__ w_raw,
                          float* __restrict__ deg,
                          float* __restrict__ ef_out) {
    int e = blockIdx.x * 256 + threadIdx.x;
    if (e >= EE) return;
    const float4* p = (const float4*)(ef + (size_t)e * 8);
    float4 a = p[0], b = p[1];
    float w = (a.x + a.y + a.z + a.w + b.x + b.y + b.z + b.w) * 0.125f;
    w_raw[e] = w;
    int r = etype[e];
    int d = eidx[EE + e];
    atomicAdd(&deg[r * NN + d], fabsf(w));
    float4* q = (float4*)(ef_out + (size_t)e * 8);
    q[0] = a; q[1] = b;
}

// ---------------------------------------------------------------------------
// 9 GEMMs [N,128]x[128,128] with V_WMMA_F32_16X16X4_F32 (wave32).
// One wave computes one 16x16 tile of one matrix j (j<8: rel_W[j] -> x_r,
// j==8: self_W + bias -> d_out).
//
// Fragment layouts per CDNA5 ISA 7.12.2:
//   A (16x4 f32):  lane L<16 : {A[M=L][k0+0], A[M=L][k0+1]}
//                  lane L>=16: {A[M=L-16][k0+2], A[M=L-16][k0+3]}
//   B (4x16 f32):  lane L<16 : {B[k0+0][N=L],   B[k0+1][N=L]}
//                  lane L>=16: {B[k0+2][N=L-16],B[k0+3][N=L-16]}
//   C/D: VGPR v -> row v (lanes 0-15) / row v+8 (lanes 16-31), col = lane%16
// ---------------------------------------------------------------------------
__global__ void gemm_wmma(const float* __restrict__ nf,
                          const float* __restrict__ relW,
                          const float* __restrict__ selfW,
                          const float* __restrict__ selfB,
                          float* __restrict__ xr,
                          float* __restrict__ out) {
    int wave = blockIdx.x * 8 + (threadIdx.x >> 5);   // 45000 waves total, exact
    int lane = threadIdx.x & 31;
    int j    = wave / (625 * 8);                      // 0..8  (which weight matrix)
    int rem  = wave % (625 * 8);
    int mt   = rem >> 3;                              // 0..624 node tile
    int nt   = rem & 7;                               // 0..7   output-col tile

    const float* Wj = (j < 8) ? (relW + (size_t)j * DD * DD) : selfW;

    int mrow  = (mt << 4) + (lane & 15);
    int col   = (nt << 4) + (lane & 15);
    int khalf = (lane >> 4) * 2;                      // 0 or 2

    const float* Abase = nf + (size_t)mrow * DD + khalf;      // 8B-aligned
    const float* Bbase = Wj + (size_t)khalf * DD + col;

    v8f c = {};
#pragma unroll 8
    for (int kt = 0; kt < 32; ++kt) {
        v2f a = *(const v2f*)(Abase + kt * 4);
        v2f b;
        b.x = Bbase[(size_t)kt * 4 * DD];
        b.y = Bbase[(size_t)kt * 4 * DD + DD];
        c = __builtin_amdgcn_wmma_f32_16x16x4_f32(
                /*neg_a=*/false, a, /*neg_b=*/false, b,
                /*c_mod=*/(short)0, c, /*reuse_a=*/false, /*reuse_b=*/false);
    }

    float bias   = (j == 8) ? selfB[col] : 0.0f;
    float* obase = (j < 8) ? (xr + (size_t)j * NN * DD) : out;
    int rbase    = (mt << 4) + ((lane >> 4) << 3);
#pragma unroll
    for (int v = 0; v < 8; ++v)
        obase[(size_t)(rbase + v) * DD + col] = c[v] + bias;
}

// ---------------------------------------------------------------------------
// One wave per edge: out[src] += (w_raw[e]/deg[r][dst]) * x_r[r][dst]
// Lane handles 4 consecutive floats: coalesced float4 gather + 4 f32 atomics.
// ---------------------------------------------------------------------------
__global__ void scatter_edges(const int*   __restrict__ eidx,
                              const int*   __restrict__ etype,
                              const float* __restrict__ w_raw,
                              const float* __restrict__ deg,
                              const float* __restrict__ xr,
                              float* __restrict__ out) {
    int e    = blockIdx.x * 8 + (threadIdx.x >> 5);   // E = 80000*8, exact
    int lane = threadIdx.x & 31;
    int s = eidx[e];
    int d = eidx[EE + e];
    int r = etype[e];
    float w = w_raw[e] / deg[r * NN + d];
    const float4* xv = (const float4*)(xr + ((size_t)r * NN + d) * DD);
    float4 v = xv[lane];
    float* o = out + (size_t)s * DD + lane * 4;
    atomicAdd(o + 0, w * v.x);
    atomicAdd(o + 1, w * v.y);
    atomicAdd(o + 2, w * v.z);
    atomicAdd(o + 3, w * v.w);
}

// ---------------------------------------------------------------------------
extern "C" void kernel_launch(void* const* d_in, const int* in_sizes, int n_in,
                              void* d_out, int out_size, void* d_ws, size_t ws_size,
                              hipStream_t stream) {
    const float* nf    = (const float*)d_in[0];   // node_features [N,128]
    const float* ef    = (const float*)d_in[1];   // edge_features [E,8]
    // d_in[2] adj_e, d_in[3] adj_v, d_in[4] T: unused (shapes only)
    const int*   eidx  = (const int*)d_in[5];     // edge_index [2,E]
    const int*   etype = (const int*)d_in[6];     // edge_type  [E]
    const float* relW  = (const float*)d_in[7];   // [R,128,128]
    const float* selfW = (const float*)d_in[8];   // [128,128]
    const float* selfB = (const float*)d_in[9];   // [128]

    float* out   = (float*)d_out;                       // [N,128]
    float* efout = (float*)d_out + (size_t)NN * DD;     // [E,8] (tuple 2nd output)

    // Workspace layout (~43.8 MB): w_raw[E] | deg[R*N] | x_r[R*N*128]
    float* w_raw = (float*)d_ws;
    float* deg   = w_raw + EE;
    float* xr    = deg + RR * NN;

    init_deg     <<<(RR * NN + 255) / 256, 256, 0, stream>>>(deg);
    edge_prep    <<<(EE + 255) / 256,      256, 0, stream>>>(ef, eidx, etype, w_raw, deg, efout);
    gemm_wmma    <<<45000 / 8,             256, 0, stream>>>(nf, relW, selfW, selfB, xr, out);
    scatter_edges<<<EE / 8,                256, 0, stream>>>(eidx, etype, w_raw, deg, xr, out);
}